// LinearLoraupdate_31894427140643
// MI455X (gfx1250) — compile-verified
//
#include <hip/hip_runtime.h>
#include <hip/hip_bf16.h>
#include <stdint.h>

// Problem constants (match reference)
#define LRS     48
#define RANK    8
#define OUTD    1024
#define IND     1024
#define BN      8              // B*N = 4*2
#define O_CHUNK 128            // o-rows per workgroup (8 waves x 16)
#define I_TILE  64             // i-columns per LDS tile
#define NT      (IND / I_TILE) // 16 tiles
#define G_STRIDE (I_TILE + 4)  // 68-float LDS row stride -> conflict-free A-frag reads
#define G_TILE_F (O_CHUNK * G_STRIDE)

typedef __attribute__((ext_vector_type(2))) float v2f;
typedef __attribute__((ext_vector_type(8))) float v8f;
typedef int v4i_vs __attribute__((vector_size(16)));   // matches builtin param type

typedef __attribute__((address_space(1))) v4i_vs* gptr_b128;
typedef __attribute__((address_space(3))) v4i_vs* lptr_b128;

#if __has_builtin(__builtin_amdgcn_global_load_async_to_lds_b128) && \
    __has_builtin(__builtin_amdgcn_s_wait_asynccnt)
#define USE_ASYNC 1
#else
#define USE_ASYNC 0
#endif

// Copy one 128 x 64 f32 gradient tile into LDS (256 threads, 8 x b128 each).
__device__ __forceinline__ void copy_tile(const float* __restrict__ gsrc, float* dst)
{
  const int t = threadIdx.x;
#pragma unroll
  for (int j = 0; j < 8; ++j) {
    int f   = t + j * 256;
    int row = f >> 4;          // 16 float4 per 64-float row
    int c4  = (f & 15) << 2;
#if USE_ASYNC
    __builtin_amdgcn_global_load_async_to_lds_b128(
        (gptr_b128)(gsrc + (size_t)row * IND + c4),
        (lptr_b128)(dst + row * G_STRIDE + c4),
        0, 0);
#else
    *(float4*)(dst + row * G_STRIDE + c4) =
        *(const float4*)(gsrc + (size_t)row * IND + c4);
#endif
  }
}

__global__ void __launch_bounds__(256)
lora_trace_kernel(const float* __restrict__ lora_A,   // [L][R][IN]
                  const float* __restrict__ lora_B,   // [L][OUT][R]
                  const float* __restrict__ grad,     // [L][B][N][OUT][IN]
                  float* __restrict__ partials)       // [L*8][BN]
{
  extern __shared__ float smem[];
  float* gbuf0 = smem;                       // 128 x 68
  float* gbuf1 = smem + G_TILE_F;            // 128 x 68
  float* aT    = smem + 2 * G_TILE_F;        // [IN][R]  (A_l transposed)
  float* bB    = aT + IND * RANK;            // [O_CHUNK][R]
  float* red   = bB + O_CHUNK * RANK;        // [8]

  const int gid = blockIdx.x;                // l*64 + bn*8 + oc
  const int oc  = gid & 7;
  const int bn  = (gid >> 3) & 7;
  const int l   = gid >> 6;
  const int o0  = oc * O_CHUNK;

  const float* gslice = grad + (((size_t)(l * BN + bn)) * OUTD + o0) * IND;
  const float* Al = lora_A + (size_t)l * RANK * IND;
  const float* Bl = lora_B + ((size_t)l * OUTD + o0) * RANK;

  // Stage A_l transposed into LDS: aT[i*R + r] = Al[r*IN + i] (coalesced reads).
  for (int idx = threadIdx.x; idx < RANK * IND; idx += 256) {
    int r = idx >> 10;            // IN = 1024
    int i = idx & (IND - 1);
    aT[i * RANK + r] = Al[idx];
  }
  // Stage B_l chunk (already [o][r] contiguous).
  for (int idx = threadIdx.x; idx < O_CHUNK * RANK; idx += 256)
    bB[idx] = Bl[idx];

  copy_tile(gslice, gbuf0);       // prologue: tile 0 in flight

  const int lane = threadIdx.x & 31;     // wave32
  const int wv   = threadIdx.x >> 5;     // 0..7
  const int mrow = lane & 15;            // M row within 16
  const int khi  = (lane >> 4) << 1;     // K sub-offset: 0 (lanes 0-15) / 2 (16-31)
  const int rcol = lane & 7;             // r column (cols 8..15 get duplicates)

  v8f acc = {0.f, 0.f, 0.f, 0.f, 0.f, 0.f, 0.f, 0.f};

  for (int it = 0; it < NT; ++it) {
    float* cur = (it & 1) ? gbuf1 : gbuf0;
    float* nxt = (it & 1) ? gbuf0 : gbuf1;

    if (it + 1 < NT) {
      copy_tile(gslice + (it + 1) * I_TILE, nxt);   // prefetch next tile
#if USE_ASYNC
      __builtin_amdgcn_s_wait_asynccnt(8);          // wait for *current* tile only
#endif
    } else {
#if USE_ASYNC
      __builtin_amdgcn_s_wait_asynccnt(0);
#endif
    }
    __syncthreads();   // current tile visible to all waves

    // GA[o_tile, r] += G_tile(16 x 64) x A_l^T(64 x 16) via 16 WMMA f32 16x16x4
    const float* arow = cur + (wv * 16 + mrow) * G_STRIDE + khi;
    const int ib = it * I_TILE + khi;
#pragma unroll
    for (int k4 = 0; k4 < I_TILE; k4 += 4) {
      v2f a = *(const v2f*)(arow + k4);                 // A-frag: 2 K-values/lane
      v2f b;
      b.x = aT[(size_t)(ib + k4) * RANK + rcol];        // B-frag row K=khi
      b.y = aT[(size_t)(ib + k4 + 1) * RANK + rcol];    // B-frag row K=khi+1
      acc = __builtin_amdgcn_wmma_f32_16x16x4_f32(
          false, a, false, b, (short)0, acc, false, false);
    }
    __syncthreads();   // all waves done with 'cur' before it is refilled
  }

  // acc VGPR v: lanes 0-15 -> (M=v, N=lane), lanes 16-31 -> (M=v+8, N=lane-16)
  float partial = 0.f;
  if ((lane & 15) < 8) {
    const int mo = wv * 16 + ((lane >> 4) << 3);
#pragma unroll
    for (int v = 0; v < 8; ++v)
      partial += acc[v] * bB[(mo + v) * RANK + rcol];
  }
#pragma unroll
  for (int off = 16; off > 0; off >>= 1)
    partial += __shfl_xor(partial, off, 32);
  if (lane == 0) red[wv] = partial;
  __syncthreads();
  if (threadIdx.x == 0) {
    float s = 0.f;
#pragma unroll
    for (int w = 0; w < 8; ++w) s += red[w];
    partials[(size_t)(l * 8 + oc) * BN + bn] = s;
  }
}

// Deterministic final reduction: fixed-order sum over 384 partials per (b,n).
__global__ void lora_trace_reduce(const float* __restrict__ partials,
                                  float* __restrict__ out)
{
  int bn = threadIdx.x;
  if (bn < BN) {
    float s = 0.f;
    for (int j = 0; j < LRS * 8; ++j)
      s += partials[(size_t)j * BN + bn];
    out[bn] = s;
  }
}

extern "C" void kernel_launch(void* const* d_in, const int* in_sizes, int n_in,
                              void* d_out, int out_size, void* d_ws, size_t ws_size,
                              hipStream_t stream)
{
  const float* lora_A = (const float*)d_in[0];
  const float* lora_B = (const float*)d_in[1];
  const float* grad   = (const float*)d_in[2];
  float* out      = (float*)d_out;
  float* partials = (float*)d_ws;     // 384*8 floats = 12 KB << ws_size

  const size_t smem =
      (size_t)(2 * G_TILE_F + IND * RANK + O_CHUNK * RANK + 8) * sizeof(float);

  dim3 grid(LRS * 8 * 8);   // (l, bn, o-chunk) = 3072 workgroups
  lora_trace_kernel<<<grid, 256, smem, stream>>>(lora_A, lora_B, grad, partials);
  lora_trace_reduce<<<1, 32, 0, stream>>>(partials, out);
}